// NeuronGraph_43336220017086
// MI455X (gfx1250) — compile-verified
//
#include <hip/hip_runtime.h>
#include <math.h>

#define NN      8192
#define N_IN    64
#define N_OUT   32
#define ROW0    (NN - N_OUT)   // 8160: first live output row
#define E_REC   65536
#define D_HIST  4

#define MATVEC_BLOCKS 8
#define EDGE_BLOCKS   64
#define THREADS       256
#define K_CHUNK       (NN / 32)   // 256 columns per wave

typedef __attribute__((ext_vector_type(2))) float v2f;
typedef __attribute__((ext_vector_type(8))) float v8f;

// Seed the 32-entry accumulator with bias of the live rows (re-done every call).
__global__ void ng_init(const float* __restrict__ bias, float* __restrict__ pre) {
    int j = threadIdx.x;
    if (j < N_OUT) pre[j] = bias[ROW0 + j];
}

__global__ void ng_compute(const float* __restrict__ obs,
                           const float* __restrict__ W,
                           const float* __restrict__ current,
                           const float* __restrict__ history,
                           const float* __restrict__ rec_w,
                           const int*   __restrict__ rec_src,
                           const int*   __restrict__ rec_dst,
                           const int*   __restrict__ rec_delay,
                           float*       __restrict__ pre) {
    if (blockIdx.x < MATVEC_BLOCKS) {
        // ---- dense part: rows [8160,8192) of W @ h_prev via V_WMMA_F32_16X16X4_F32
        const int wave  = blockIdx.x * (THREADS / 32) + (threadIdx.x >> 5); // 0..63
        const int lane  = threadIdx.x & 31;
        const int half  = lane >> 4;      // 0: K={0,1}, 1: K={2,3}
        const int lrow  = lane & 15;      // A-matrix row within tile
        const int mtile = wave & 1;       // which 16-row tile
        const int kbase = (wave >> 1) * K_CHUNK;
        const int row   = ROW0 + mtile * 16 + lrow;
        const float* Wrow = W + (size_t)row * NN;

        v8f acc = {};
        for (int k = kbase; k < kbase + K_CHUNK; k += 4) {
            const int kk = k + 2 * half;           // this lane's K pair
            v2f a, b;
            a.x = Wrow[kk];
            a.y = Wrow[kk + 1];
            // h_prev: inputs take obs, the rest keep previous activation
            b.x = (kk     < N_IN) ? obs[kk]     : current[kk];
            b.y = (kk + 1 < N_IN) ? obs[kk + 1] : current[kk + 1];
            // D = A(16x4) x B(4x16) + C ; all 16 B-columns identical -> every
            // D column holds the partial dot product for its row.
            acc = __builtin_amdgcn_wmma_f32_16x16x4_f32(
                false, a, false, b, (short)0, acc, false, false);
        }
        // D layout: VGPR r, lane half h -> row (r + 8*h); columns all equal,
        // so lanes 0 and 16 each publish 8 rows.
        if (lrow == 0) {
#pragma unroll
            for (int r = 0; r < 8; ++r)
                atomicAdd(&pre[mtile * 16 + r + 8 * half], acc[r]);
        }
    } else {
        // ---- sparse recurrent edges: only dst in the live 32 rows matter
        const int tid    = (blockIdx.x - MATVEC_BLOCKS) * THREADS + threadIdx.x;
        const int stride = EDGE_BLOCKS * THREADS;  // 16384
        for (int e = tid; e < E_REC; e += stride) {
            const int dst = rec_dst[e];
            if (dst >= ROW0) {
                const float v =
                    rec_w[e] * history[rec_src[e] * D_HIST + rec_delay[e]];
                atomicAdd(&pre[dst - ROW0], v);
            }
        }
    }
}

__global__ void ng_finish(const float* __restrict__ pre, float* __restrict__ out) {
    int j = threadIdx.x;
    if (j < N_OUT) out[j] = tanhf(pre[j]);
}

extern "C" void kernel_launch(void* const* d_in, const int* in_sizes, int n_in,
                              void* d_out, int out_size, void* d_ws, size_t ws_size,
                              hipStream_t stream) {
    (void)in_sizes; (void)n_in; (void)out_size; (void)ws_size;
    const float* obs       = (const float*)d_in[0];
    const float* W         = (const float*)d_in[1];
    const float* bias      = (const float*)d_in[2];
    const float* current   = (const float*)d_in[3];
    const float* history   = (const float*)d_in[4];
    const float* rec_w     = (const float*)d_in[5];
    const int*   rec_src   = (const int*)d_in[6];
    const int*   rec_dst   = (const int*)d_in[7];
    const int*   rec_delay = (const int*)d_in[8];
    float* out = (float*)d_out;
    float* pre = (float*)d_ws;   // 32 floats of scratch

    ng_init<<<1, 32, 0, stream>>>(bias, pre);
    ng_compute<<<MATVEC_BLOCKS + EDGE_BLOCKS, THREADS, 0, stream>>>(
        obs, W, current, history, rec_w, rec_src, rec_dst, rec_delay, pre);
    ng_finish<<<1, 32, 0, stream>>>(pre, out);
}